// GaussianRasterizer_15221364097782
// MI455X (gfx1250) — compile-verified
//
#include <hip/hip_runtime.h>
#include <math.h>

typedef float v2f __attribute__((ext_vector_type(2)));
typedef float v8f __attribute__((ext_vector_type(8)));

#define IMG_H 192
#define IMG_W 192
#define NPIX (IMG_H * IMG_W)
#define FOCAL 192.0f            // W / (2*tanfovx) = 192/1.0
#define TANLIM 0.65f            // 1.3 * tanfov
#define C_NEAR 0.2f
#define C_ALPHA_MIN (1.0f / 255.0f)
#define C_T_EPS 1.0e-4f

// ---------------------------------------------------------------------------
// Kernel 1: per-gaussian preprocessing + stable depth sort + scatter of
// WMMA-ready feature vectors into workspace (sorted order).
// One block of 256 threads (P == 256).
// ---------------------------------------------------------------------------
__global__ __launch_bounds__(256) void gs_preprocess(
    const float* __restrict__ means3D,
    const float* __restrict__ opacities,
    const float* __restrict__ colors,
    const float* __restrict__ scales,
    const float* __restrict__ rotations,
    const float* __restrict__ viewm,
    const float* __restrict__ projm,
    float* __restrict__ featWS,   // [P][8] sorted gaussian features (A-matrix rows)
    float* __restrict__ opWS,     // [P]    sorted opacity (0 if invalid)
    float* __restrict__ colWS,    // [P][4] sorted color (padded)
    int*   __restrict__ radiiOut, // [P]    radii in ORIGINAL order
    int P)
{
  __shared__ float sdepth[256];
  const int i = threadIdx.x;

  float m0 = 0.f, m1 = 0.f, m2 = 0.f;
  float q0 = 1.f, q1 = 0.f, q2 = 0.f, q3 = 0.f;
  float s0 = 0.f, s1 = 0.f, s2v = 0.f;
  float opv = 0.f, col0 = 0.f, col1 = 0.f, col2 = 0.f;
  if (i < P) {
    m0 = means3D[i * 3 + 0]; m1 = means3D[i * 3 + 1]; m2 = means3D[i * 3 + 2];
    q0 = rotations[i * 4 + 0]; q1 = rotations[i * 4 + 1];
    q2 = rotations[i * 4 + 2]; q3 = rotations[i * 4 + 3];
    s0 = scales[i * 3 + 0]; s1 = scales[i * 3 + 1]; s2v = scales[i * 3 + 2];
    opv = opacities[i];
    col0 = colors[i * 3 + 0]; col1 = colors[i * 3 + 1]; col2 = colors[i * 3 + 2];
  }

  // p_view = m @ V[:3,:3] + V[3,:3]   (row-vector convention, row-major V)
  float pv0 = m0 * viewm[0] + m1 * viewm[4] + m2 * viewm[8]  + viewm[12];
  float pv1 = m0 * viewm[1] + m1 * viewm[5] + m2 * viewm[9]  + viewm[13];
  float pv2 = m0 * viewm[2] + m1 * viewm[6] + m2 * viewm[10] + viewm[14];
  // p_hom = [m,1] @ proj
  float ph0 = m0 * projm[0] + m1 * projm[4] + m2 * projm[8]  + projm[12];
  float ph1 = m0 * projm[1] + m1 * projm[5] + m2 * projm[9]  + projm[13];
  float ph3 = m0 * projm[3] + m1 * projm[7] + m2 * projm[11] + projm[15];
  float invw = 1.0f / (ph3 + 1e-7f);
  float pp0 = ph0 * invw, pp1 = ph1 * invw;
  float depth = pv2;
  bool infr = depth > C_NEAR;

  // quaternion -> rotation
  float qn = 1.0f / sqrtf(q0 * q0 + q1 * q1 + q2 * q2 + q3 * q3);
  float r = q0 * qn, x = q1 * qn, y = q2 * qn, z = q3 * qn;
  float R00 = 1.f - 2.f * (y * y + z * z), R01 = 2.f * (x * y - r * z), R02 = 2.f * (x * z + r * y);
  float R10 = 2.f * (x * y + r * z), R11 = 1.f - 2.f * (x * x + z * z), R12 = 2.f * (y * z - r * x);
  float R20 = 2.f * (x * z - r * y), R21 = 2.f * (y * z + r * x), R22 = 1.f - 2.f * (x * x + y * y);
  float e0 = s0 * s0, e1 = s1 * s1, e2 = s2v * s2v;   // SCALE_MOD = 1
  // Sigma = R diag(e) R^T (symmetric)
  float S00 = R00 * R00 * e0 + R01 * R01 * e1 + R02 * R02 * e2;
  float S01 = R00 * R10 * e0 + R01 * R11 * e1 + R02 * R12 * e2;
  float S02 = R00 * R20 * e0 + R01 * R21 * e1 + R02 * R22 * e2;
  float S11 = R10 * R10 * e0 + R11 * R11 * e1 + R12 * R12 * e2;
  float S12 = R10 * R20 * e0 + R11 * R21 * e1 + R12 * R22 * e2;
  float S22 = R20 * R20 * e0 + R21 * R21 * e1 + R22 * R22 * e2;

  float tz = infr ? depth : 1.0f;
  float itz = 1.0f / tz;
  float tx = fminf(fmaxf(pv0 * itz, -TANLIM), TANLIM) * tz;
  float ty = fminf(fmaxf(pv1 * itz, -TANLIM), TANLIM) * tz;
  float J00 = FOCAL * itz, J02 = -FOCAL * tx * itz * itz;
  float J11 = FOCAL * itz, J12 = -FOCAL * ty * itz * itz;
  // Wr = V[:3,:3]^T : Wr[k][j] = viewm[j*4+k];  Tm = J @ Wr
  float T00 = J00 * viewm[0] + J02 * viewm[2];
  float T01 = J00 * viewm[4] + J02 * viewm[6];
  float T02 = J00 * viewm[8] + J02 * viewm[10];
  float T10 = J11 * viewm[1] + J12 * viewm[2];
  float T11 = J11 * viewm[5] + J12 * viewm[6];
  float T12 = J11 * viewm[9] + J12 * viewm[10];
  // cov2d = Tm Sigma Tm^T
  float t00 = T00 * S00 + T01 * S01 + T02 * S02;
  float t01 = T00 * S01 + T01 * S11 + T02 * S12;
  float t02 = T00 * S02 + T01 * S12 + T02 * S22;
  float t10 = T10 * S00 + T11 * S01 + T12 * S02;
  float t11 = T10 * S01 + T11 * S11 + T12 * S12;
  float t12 = T10 * S02 + T11 * S12 + T12 * S22;
  float cov00 = t00 * T00 + t01 * T01 + t02 * T02;
  float cov01 = t00 * T10 + t01 * T11 + t02 * T12;
  float cov11 = t10 * T10 + t11 * T11 + t12 * T12;

  float c00 = cov00 + 0.3f, c11 = cov11 + 0.3f, c01 = cov01;
  float det = c00 * c11 - c01 * c01;
  bool valid = infr && (det > 0.f);
  float dets = (det > 0.f) ? det : 1.0f;
  float idet = 1.0f / dets;
  float conA = c11 * idet, conB = -c01 * idet, conC = c00 * idet;
  float mid = 0.5f * (c00 + c11);
  float lam1 = mid + sqrtf(fmaxf(0.1f, mid * mid - det));
  int radii = valid ? (int)ceilf(3.0f * sqrtf(lam1)) : 0;
  float px = ((pp0 + 1.0f) * (float)IMG_W - 1.0f) * 0.5f;
  float py = ((pp1 + 1.0f) * (float)IMG_H - 1.0f) * 0.5f;

  // stable rank sort by depth
  if (i < P) sdepth[i] = depth;
  __syncthreads();
  if (i < P) {
    int rank = 0;
    for (int j = 0; j < P; ++j) {
      float dj = sdepth[j];
      rank += (dj < depth) || ((dj == depth) && (j < i)) ? 1 : 0;
    }
    // Per-gaussian WMMA feature row: power[g][pix] = f . (x^2,y^2,xy,x,y,1,0,0)
    float f[8];
    if (valid) {
      f[0] = -0.5f * conA;
      f[1] = -0.5f * conC;
      f[2] = -conB;
      f[3] = conA * px + conB * py;
      f[4] = conC * py + conB * px;
      f[5] = -0.5f * (conA * px * px + conC * py * py) - conB * px * py;
      f[6] = 0.f; f[7] = 0.f;
    } else {
      f[0] = f[1] = f[2] = f[3] = f[4] = 0.f;
      f[5] = -1.0e4f;      // power = -1e4 -> alpha 0 -> skipped
      f[6] = f[7] = 0.f;
    }
#pragma unroll
    for (int k = 0; k < 8; ++k) featWS[rank * 8 + k] = f[k];
    opWS[rank] = valid ? opv : 0.0f;
    colWS[rank * 4 + 0] = col0;
    colWS[rank * 4 + 1] = col1;
    colWS[rank * 4 + 2] = col2;
    colWS[rank * 4 + 3] = 0.f;
    radiiOut[i] = radii;   // original order
  }
}

// ---------------------------------------------------------------------------
// Kernel 2: rank-8 GEMM power evaluation via V_WMMA_F32_16X16X4_F32 + per-pixel
// front-to-back compositing. 256 threads = 8 waves; each wave owns 32 pixels.
// ---------------------------------------------------------------------------
__global__ __launch_bounds__(256) void gs_render(
    const float* __restrict__ featWS,
    const float* __restrict__ opWS,
    const float* __restrict__ colWS,
    const float* __restrict__ bg,
    float* __restrict__ outImg,
    int ntiles)
{
  __shared__ float pw[8][16][32];   // per-wave 16-gaussian x 32-pixel power tile
  const int lane = threadIdx.x & 31;
  const int wave = threadIdx.x >> 5;
  const int base = blockIdx.x * 256 + wave * 32;
  const int half = lane >> 4;       // 0: K rows {0,1}/{4,5}; 1: K rows {2,3}/{6,7}
  const int nl = lane & 15;

  __builtin_prefetch(featWS, 0, 1); // global_prefetch of gaussian features

  // per-pixel feature vectors (B-matrix columns), pixels base+nl and base+16+nl
  float flo[8], fhi[8];
  {
    int plo = base + nl;
    float xl = (float)(plo % IMG_W), yl = (float)(plo / IMG_W);
    flo[0] = xl * xl; flo[1] = yl * yl; flo[2] = xl * yl;
    flo[3] = xl; flo[4] = yl; flo[5] = 1.f; flo[6] = 0.f; flo[7] = 0.f;
    int phi = base + 16 + nl;
    float xh = (float)(phi % IMG_W), yh = (float)(phi / IMG_W);
    fhi[0] = xh * xh; fhi[1] = yh * yh; fhi[2] = xh * yh;
    fhi[3] = xh; fhi[4] = yh; fhi[5] = 1.f; fhi[6] = 0.f; fhi[7] = 0.f;
  }
  v2f b0lo = { flo[2 * half], flo[2 * half + 1] };
  v2f b1lo = { flo[4 + 2 * half], flo[5 + 2 * half] };
  v2f b0hi = { fhi[2 * half], fhi[2 * half + 1] };
  v2f b1hi = { fhi[4 + 2 * half], fhi[5 + 2 * half] };

  float Tv = 1.0f;
  float aR = 0.f, aG = 0.f, aB = 0.f;
  bool done = false;

  for (int t = 0; t < ntiles; ++t) {
    const int g0 = t * 16;
    // A-matrix (16 gaussians x 8): lane holds gaussian g0+nl, K pair per `half`
    const float* fr = featWS + (size_t)(g0 + nl) * 8;
    v2f a0 = { fr[2 * half], fr[2 * half + 1] };
    v2f a1 = { fr[4 + 2 * half], fr[5 + 2 * half] };

    v8f clo = {};
    v8f chi = {};
    clo = __builtin_amdgcn_wmma_f32_16x16x4_f32(false, a0, false, b0lo, (short)0, clo, false, false);
    clo = __builtin_amdgcn_wmma_f32_16x16x4_f32(false, a1, false, b1lo, (short)0, clo, false, false);
    chi = __builtin_amdgcn_wmma_f32_16x16x4_f32(false, a0, false, b0hi, (short)0, chi, false, false);
    chi = __builtin_amdgcn_wmma_f32_16x16x4_f32(false, a1, false, b1hi, (short)0, chi, false, false);

    // Stage tile in LDS: lane l, VGPR v holds D[M=v+8*half][N=nl]
#pragma unroll
    for (int v = 0; v < 8; ++v) {
      pw[wave][v + 8 * half][nl]      = clo[v];
      pw[wave][v + 8 * half][16 + nl] = chi[v];
    }
    // Same-wave LDS ops are in-order; compiler inserts s_wait_dscnt for the reads.

    // Sequential front-to-back compositing for this 16-gaussian slab
    for (int g = 0; g < 16; ++g) {
      float pwr = pw[wave][g][lane];
      if (!done && pwr <= 0.0f) {
        float alpha = fminf(0.99f, opWS[g0 + g] * __expf(pwr));
        if (alpha >= C_ALPHA_MIN) {
          float tst = Tv * (1.0f - alpha);
          if (tst < C_T_EPS) {
            done = true;            // stop-triggering gaussian contributes nothing
          } else {
            float wgt = alpha * Tv;
            aR += wgt * colWS[(g0 + g) * 4 + 0];
            aG += wgt * colWS[(g0 + g) * 4 + 1];
            aB += wgt * colWS[(g0 + g) * 4 + 2];
            Tv = tst;
          }
        }
      }
    }
    if (__builtin_amdgcn_ballot_w32(!done) == 0u) break;  // wave-uniform early exit
  }

  const int myPix = base + lane;
  outImg[0 * NPIX + myPix] = aR + bg[0] * Tv;
  outImg[1 * NPIX + myPix] = aG + bg[1] * Tv;
  outImg[2 * NPIX + myPix] = aB + bg[2] * Tv;
}

// ---------------------------------------------------------------------------
extern "C" void kernel_launch(void* const* d_in, const int* in_sizes, int n_in,
                              void* d_out, int out_size, void* d_ws, size_t ws_size,
                              hipStream_t stream) {
  const float* means3D = (const float*)d_in[0];
  /* d_in[1] = means2D (unused by reference) */
  const float* opac    = (const float*)d_in[2];
  const float* colors  = (const float*)d_in[3];
  const float* scales  = (const float*)d_in[4];
  const float* rots    = (const float*)d_in[5];
  const float* bg      = (const float*)d_in[6];
  const float* viewm   = (const float*)d_in[7];
  const float* projm   = (const float*)d_in[8];
  const int P = in_sizes[0] / 3;   // 256

  float* featWS = (float*)d_ws;                    // P*8 floats
  float* opWS   = featWS + (size_t)P * 8;          // P floats
  float* colWS  = opWS + P;                        // P*4 floats
  float* img    = (float*)d_out;                   // 3*192*192 floats
  int*   radii  = (int*)d_out + 3 * NPIX;          // P int32 (bit-pattern concat)

  gs_preprocess<<<1, 256, 0, stream>>>(means3D, opac, colors, scales, rots,
                                       viewm, projm, featWS, opWS, colWS, radii, P);
  const int ntiles = (P + 15) / 16;
  gs_render<<<NPIX / 256, 256, 0, stream>>>(featWS, opWS, colWS, bg, img, ntiles);
}